// MSBlock_58944131170357
// MI455X (gfx1250) — compile-verified
//
#include <hip/hip_runtime.h>

// ---------------------------------------------------------------------------
// MSBlock for MI455X (gfx1250, wave32). fp32 WMMA 16x16x4 for all channel-
// mixing GEMMs (memory-bound problem -> keep full precision). Tail 1x1 convs
// + branch prefix/suffix sums are folded into one precombined 512x1024 GEMM.
// Global->LDS staging uses the CDNA5 Tensor Data Mover (tensor_load_to_lds)
// when the builtin is available, else global_load_async_to_lds_* inline asm.
// ---------------------------------------------------------------------------

#if defined(__has_builtin)
#if __has_builtin(__builtin_amdgcn_tensor_load_to_lds)
#define MS_TDM 1
#endif
#endif
#ifndef MS_TDM
#define MS_TDM 0
#endif

namespace {

constexpr int Bn  = 64;
constexpr int Cn  = 512;
constexpr int Nn  = 1024;
constexpr int BCn = 128;

typedef __attribute__((ext_vector_type(2))) float v2f;
typedef __attribute__((ext_vector_type(8))) float v8f;
typedef unsigned int u32x4 __attribute__((ext_vector_type(4)));
typedef int i32x4 __attribute__((ext_vector_type(4)));
typedef int i32x8 __attribute__((ext_vector_type(8)));

__constant__ float c_dec_lo[8] = {
    -0.010597401784997278f, 0.032883011666982945f, 0.030841381835986965f,
    -0.18703481171888114f,  -0.02798376941698385f, 0.6308807679295904f,
    0.7148465705525415f,    0.23037781330885523f};
__constant__ float c_dec_hi[8] = {
    -0.23037781330885523f,  0.7148465705525415f,  -0.6308807679295904f,
    -0.02798376941698385f,  0.18703481171888114f,  0.030841381835986965f,
    -0.032883011666982945f, -0.010597401784997278f};

#define LDA 20   // padded LDS stride for A (128x16 tile): 16 dwords + 4 pad
#define LDB 132  // padded LDS stride for B (16x128 tile): 128 dwords + 4 pad

// Low 32 bits of a flat pointer into LDS == workgroup-relative LDS byte addr
// (ISA 10.2: LDS aperture).
__device__ __forceinline__ unsigned lds_addr_of(const void* p) {
  return (unsigned)(unsigned long long)p;
}

// --- CDNA5 async global->LDS copies (tracked by ASYNCcnt) -------------------
__device__ __forceinline__ void async_b128(unsigned lds, const void* g) {
  asm volatile("global_load_async_to_lds_b128 %0, %1, off"
               :: "v"(lds), "v"((unsigned long long)g) : "memory");
}
__device__ __forceinline__ void async_b32(unsigned lds, const void* g) {
  asm volatile("global_load_async_to_lds_b32 %0, %1, off"
               :: "v"(lds), "v"((unsigned long long)g) : "memory");
}
__device__ __forceinline__ void async_wait() {
  asm volatile("s_wait_asynccnt 0x0" ::: "memory");
}

// --- CDNA5 Tensor Data Mover: 2D tile load with LDS padding -----------------
// Loads a tile_h x tile_w (dwords) tile from a row-major tensor of width
// tensor_w (row stride stride_w dwords) into LDS, inserting pad dwords so the
// LDS row pitch matches our padded layouts. Issued once per wave (EXEC
// ignored by TDM) -> call from wave 0 only. Tracked by TENSORcnt.
__device__ __forceinline__ void tdm_load_2d(unsigned lds_byte, const void* gp,
                                            int tile_w, int tile_h,
                                            int tensor_w, int tensor_h,
                                            int stride_w,
                                            int pad_int_code, int pad_amt_code) {
#if MS_TDM
  const unsigned long long ga = (unsigned long long)gp;
  u32x4 g0;
  g0[0] = 1u;                                  // count=1, user descriptor
  g0[1] = lds_byte;                            // lds_addr [63:32]
  g0[2] = (unsigned)ga;                        // global_addr[31:0]
  g0[3] = (unsigned)((ga >> 32) & 0x1FFFFFFull) | 0x80000000u;  // [56:32]|type=2
  i32x8 g1;
  g1[0] = (2 << 16)                            // data_size = 4 bytes
        | (1 << 20)                            // pad_enable
        | (pad_int_code << 22)                 // dwords before pad (3=16, 6=128)
        | (pad_amt_code << 25);                // pad dwords - 1 (3 = 4 dwords)
  g1[1] = (tensor_w & 0xFFFF) << 16;           // tensor_dim0[15:0] @ bits 63:48
  g1[2] = (int)(((unsigned)tensor_w >> 16) | ((unsigned)(tensor_h & 0xFFFF) << 16));
  g1[3] = (int)(((unsigned)tensor_h >> 16) | ((unsigned)tile_w << 16));  // tile_dim0
  g1[4] = tile_h & 0xFFFF;                     // tile_dim1 (tile_dim2 = 0)
  g1[5] = stride_w;                            // tensor_dim0_stride[31:0]
  g1[6] = 0;                                   // stride hi, dim1_stride
  g1[7] = 0;
  i32x4 z4 = {0, 0, 0, 0};
#if __clang_major__ >= 23
  i32x8 z8 = {0, 0, 0, 0, 0, 0, 0, 0};
  __builtin_amdgcn_tensor_load_to_lds(g0, g1, z4, z4, z8, 0);
#else
  __builtin_amdgcn_tensor_load_to_lds(g0, g1, z4, z4, 0);
#endif
#else
  (void)lds_byte; (void)gp; (void)tile_w; (void)tile_h; (void)tensor_w;
  (void)tensor_h; (void)stride_w; (void)pad_int_code; (void)pad_amt_code;
#endif
}
__device__ __forceinline__ void tdm_wait() {
#if MS_TDM
  __builtin_amdgcn_s_wait_tensorcnt(0);
#endif
}

// Wave-level fp32 WMMA over one K=16 LDS tile. WG = 256 threads = 8 waves.
// Wave (wm, wn) owns a 32x64 sub-tile: 2 (M) x 4 (N) accumulators of 16x16.
// A frag (16x4 f32): lanes 0-15 hold K={k,k+1}, lanes 16-31 hold K={k+2,k+3}.
// B frag (4x16 f32): VGPR0 = rows {k, k+2}, VGPR1 = rows {k+1, k+3}.
__device__ __forceinline__ void wmma_accum_16(const float* Alds, const float* Blds,
                                              int lane, int wave_m, int wave_n,
                                              v8f acc[2][4]) {
  const int half = lane >> 4;
  const int l16  = lane & 15;
#pragma unroll
  for (int kk = 0; kk < 16; kk += 4) {
    const int ka = kk + half * 2;
    v2f a[2];
#pragma unroll
    for (int mi = 0; mi < 2; ++mi) {
      const int row = wave_m * 32 + mi * 16 + l16;
      a[mi].x = Alds[row * LDA + ka];
      a[mi].y = Alds[row * LDA + ka + 1];
    }
    v2f bf[4];
#pragma unroll
    for (int ni = 0; ni < 4; ++ni) {
      const int col = wave_n * 64 + ni * 16 + l16;
      bf[ni].x = Blds[(ka)*LDB + col];
      bf[ni].y = Blds[(ka + 1) * LDB + col];
    }
#pragma unroll
    for (int mi = 0; mi < 2; ++mi)
#pragma unroll
      for (int ni = 0; ni < 4; ++ni)
        acc[mi][ni] = __builtin_amdgcn_wmma_f32_16x16x4_f32(
            false, a[mi], false, bf[ni], (short)0, acc[mi][ni], false, false);
  }
}

}  // namespace

// --- K0: combine w_fus/w_pf + cumulative branch sums into G (512x1024) ------
__global__ __launch_bounds__(256) void k_combine(
    const float* __restrict__ w_fus, const float* __restrict__ w_pf,
    const float* __restrict__ b_fus, const float* __restrict__ b_pf,
    float* __restrict__ G, float* __restrict__ biasc) {
  const int idx = blockIdx.x * 256 + threadIdx.x;  // 512*1024
  const int r  = idx >> 10;
  const int cp = idx & 1023;
  float acc = 0.f;
  if (cp < Cn) {  // shorts part: G_S_i = sum_{bb>=i} Wfus_bb @ Wpf[:, :128]
    const int i = cp >> 7, k = cp & 127;
    for (int bb = i; bb < 4; ++bb)
      for (int j = 0; j < BCn; ++j)
        acc += w_fus[(size_t)r * Cn + bb * BCn + j] * w_pf[(size_t)j * 256 + k];
  } else {        // longs part: G_L_i = sum_{bb<=i} Wfus_bb @ Wpf[:, 128:]
    const int i = (cp - Cn) >> 7, k = cp & 127;
    for (int bb = 0; bb <= i; ++bb)
      for (int j = 0; j < BCn; ++j)
        acc += w_fus[(size_t)r * Cn + bb * BCn + j] * w_pf[(size_t)j * 256 + 128 + k];
  }
  G[(size_t)r * 1024 + cp] = acc;
  if (cp == 0) {
    float bacc = b_fus[r];
    for (int q = 0; q < Cn; ++q)
      bacc += w_fus[(size_t)r * Cn + q] * b_pf[q & 127];
    biasc[r] = bacc;
  }
}

// --- K1: H = W_init(512x512) @ X_b + b_init  (WMMA GEMM, TDM staging) -------
__global__ __launch_bounds__(256) void k_gemm_init(
    const float* __restrict__ X, const float* __restrict__ W,
    const float* __restrict__ bias, float* __restrict__ H) {
  __shared__ __align__(16) float Alds[128 * LDA];
  __shared__ __align__(16) float Blds[16 * LDB];
  const int tid = threadIdx.x;
  const int lane = tid & 31, wave = tid >> 5;
  const int wave_m = wave & 3, wave_n = wave >> 2;
  const int n0 = blockIdx.x * 128;
  const int m0 = blockIdx.y * 128;
  const int b  = blockIdx.z;
  const float* Xb = X + (size_t)b * Cn * Nn;
  v8f acc[2][4] = {};
  for (int k0 = 0; k0 < Cn; k0 += 16) {
    if (MS_TDM) {
      if (tid < 32) {  // one TDM descriptor pair per workgroup tile
        tdm_load_2d(lds_addr_of(Alds), W + (size_t)m0 * Cn + k0,
                    16, 128, Cn, Cn, Cn, /*pad per 16*/ 3, 3);
        tdm_load_2d(lds_addr_of(Blds), Xb + (size_t)k0 * Nn + n0,
                    128, 16, Nn, Cn, Nn, /*pad per 128*/ 6, 3);
        tdm_wait();
      }
    } else {
#pragma unroll
      for (int i = 0; i < 2; ++i) {  // A tile: 128x16
        const int idx = tid + i * 256;
        const int row = idx >> 2;
        const int c4  = (idx & 3) << 2;
        async_b128(lds_addr_of(&Alds[row * LDA + c4]),
                   W + (size_t)(m0 + row) * Cn + k0 + c4);
      }
#pragma unroll
      for (int i = 0; i < 2; ++i) {  // B tile: 16x128
        const int idx = tid + i * 256;
        const int row = idx >> 5;
        const int c4  = (idx & 31) << 2;
        async_b128(lds_addr_of(&Blds[row * LDB + c4]),
                   Xb + (size_t)(k0 + row) * Nn + n0 + c4);
      }
      async_wait();
    }
    __syncthreads();
    wmma_accum_16(Alds, Blds, lane, wave_m, wave_n, acc);
    __syncthreads();
  }
  const int half = lane >> 4, l16 = lane & 15;
  float* Hb = H + (size_t)b * Cn * Nn;
#pragma unroll
  for (int mi = 0; mi < 2; ++mi)
#pragma unroll
    for (int ni = 0; ni < 4; ++ni)
#pragma unroll
      for (int r = 0; r < 8; ++r) {
        const int m = m0 + wave_m * 32 + mi * 16 + r + half * 8;
        const int n = n0 + wave_n * 64 + ni * 16 + l16;
        Hb[(size_t)m * Nn + n] = acc[mi][ni][r] + bias[m];
      }
}

// --- K2: branch K-tap convs as tap-decomposed WMMA GEMMs --------------------
__global__ __launch_bounds__(256) void k_branch_conv(
    const float* __restrict__ H,
    const float* __restrict__ w1, const float* __restrict__ w2,
    const float* __restrict__ w3, const float* __restrict__ w4,
    const float* __restrict__ cb1, const float* __restrict__ cb2,
    const float* __restrict__ cb3, const float* __restrict__ cb4,
    float* __restrict__ Y) {
  __shared__ __align__(16) float Alds[128 * LDA];
  __shared__ __align__(16) float Blds[16 * LDB];
  const int tid = threadIdx.x;
  const int lane = tid & 31, wave = tid >> 5;
  const int wave_m = wave & 3, wave_n = wave >> 2;
  const int n0 = blockIdx.x * 128;
  const int br = blockIdx.y;
  const int b  = blockIdx.z;
  const int CB = br * BCn;
  const float* wb = (br == 0) ? w1 : (br == 1) ? w2 : (br == 2) ? w3 : w4;
  const float* cb = (br == 0) ? cb1 : (br == 1) ? cb2 : (br == 2) ? cb3 : cb4;
  const int ki  = 3 + 2 * br;  // KS = {3,5,7,9}
  const int pad = ki >> 1;
  const float* Hb = H + ((size_t)b * Cn + CB) * Nn;
  v8f acc[2][4] = {};
  for (int t = 0; t < ki; ++t) {
    const int sh = t - pad;
    const bool interior = (n0 + sh >= 0) && (n0 + 127 + sh < Nn);
    for (int k0 = 0; k0 < BCn; k0 += 16) {
#pragma unroll
      for (int i = 0; i < 8; ++i) {  // A: weights[:, k0:k0+16, t] (stride ki)
        const int idx = tid + i * 256;
        const int row = idx >> 4;
        const int cc  = idx & 15;
        async_b32(lds_addr_of(&Alds[row * LDA + cc]),
                  wb + ((size_t)row * BCn + k0 + cc) * ki + t);
      }
      if (interior) {
#pragma unroll
        for (int i = 0; i < 8; ++i) {  // B: shifted H window (fully in range)
          const int idx = tid + i * 256;
          const int row = idx >> 7;
          const int col = idx & 127;
          async_b32(lds_addr_of(&Blds[row * LDB + col]),
                    Hb + (size_t)(k0 + row) * Nn + n0 + col + sh);
        }
      } else {
#pragma unroll
        for (int i = 0; i < 8; ++i) {  // boundary tile: zero-padded edges
          const int idx = tid + i * 256;
          const int row = idx >> 7;
          const int col = idx & 127;
          const int nn  = n0 + col + sh;
          float v = 0.0f;
          if (nn >= 0 && nn < Nn) v = Hb[(size_t)(k0 + row) * Nn + nn];
          Blds[row * LDB + col] = v;
        }
      }
      async_wait();
      __syncthreads();
      wmma_accum_16(Alds, Blds, lane, wave_m, wave_n, acc);
      __syncthreads();
    }
  }
  const int half = lane >> 4, l16 = lane & 15;
  float* Yb = Y + ((size_t)b * Cn + CB) * Nn;
#pragma unroll
  for (int mi = 0; mi < 2; ++mi)
#pragma unroll
    for (int ni = 0; ni < 4; ++ni)
#pragma unroll
      for (int r = 0; r < 8; ++r) {
        const int m = wave_m * 32 + mi * 16 + r + half * 8;
        const int n = n0 + wave_n * 64 + ni * 16 + l16;
        Yb[(size_t)m * Nn + n] = acc[mi][ni][r] + cb[m];
      }
}

// --- K3: per-channel BN statistics over (B, N) ------------------------------
__global__ __launch_bounds__(256) void k_stats(const float* __restrict__ Y,
                                               float* __restrict__ stats) {
  __shared__ float rs[256];
  __shared__ float rq[256];
  const int c = blockIdx.x;
  const int tid = threadIdx.x;
  float s = 0.f, q = 0.f;
  for (int i = tid; i < Bn * Nn; i += 256) {
    const int b = i >> 10;
    const int n = i & (Nn - 1);
    const float v = Y[((size_t)b * Cn + c) * Nn + n];
    s += v;
    q += v * v;
  }
  rs[tid] = s;
  rq[tid] = q;
  __syncthreads();
  for (int off = 128; off > 0; off >>= 1) {
    if (tid < off) {
      rs[tid] += rs[tid + off];
      rq[tid] += rq[tid + off];
    }
    __syncthreads();
  }
  if (tid == 0) {
    const float invn = 1.0f / (float)(Bn * Nn);
    const float mean = rs[0] * invn;
    const float var  = rq[0] * invn - mean * mean;
    stats[c]      = mean;
    stats[Cn + c] = rsqrtf(var + 1e-5f);
  }
}

// --- K4: BN + ReLU + 2-level db4 DWT + dual IDWT (decouple), per (b,c) row --
__global__ __launch_bounds__(128) void k_bn_wavelet(
    const float* __restrict__ Y, const float* __restrict__ stats,
    const float* __restrict__ g1, const float* __restrict__ g2,
    const float* __restrict__ g3, const float* __restrict__ g4,
    const float* __restrict__ e1, const float* __restrict__ e2,
    const float* __restrict__ e3, const float* __restrict__ e4,
    float* __restrict__ S, float* __restrict__ L) {
  __shared__ float a0[1024];
  __shared__ float A1[516], D1[516];
  __shared__ float A2[262], D2[262];
  __shared__ float la2[516], sa2[516];
  const int c = blockIdx.x;
  const int b = blockIdx.y;
  const int br = c >> 7, cl = c & 127;
  const float* gp = (br == 0) ? g1 : (br == 1) ? g2 : (br == 2) ? g3 : g4;
  const float* ep = (br == 0) ? e1 : (br == 1) ? e2 : (br == 2) ? e3 : e4;
  const float mean = stats[c];
  const float gmul = stats[Cn + c] * gp[cl];
  const float beta = ep[cl];
  const int tid = threadIdx.x;
  const float* yr = Y + ((size_t)b * Cn + c) * Nn;

  for (int n = tid; n < Nn; n += 128) {  // BN + ReLU
    const float v = (yr[n] - mean) * gmul + beta;
    a0[n] = v > 0.f ? v : 0.f;
  }
  __syncthreads();

  // DWT level 1: n=1024 -> 515, symmetric pad 6/6, stride 2, flipped taps
  for (int o = tid; o < 515; o += 128) {
    float lo = 0.f, hi = 0.f;
#pragma unroll
    for (int j = 0; j < 8; ++j) {
      int q = 2 * o + j - 6;
      const int qi = (q < 0) ? (-1 - q) : ((q >= 1024) ? (2047 - q) : q);
      const float v = a0[qi];
      lo += c_dec_lo[7 - j] * v;
      hi += c_dec_hi[7 - j] * v;
    }
    A1[o] = lo;
    D1[o] = hi;
  }
  __syncthreads();

  // DWT level 2: n=515 -> 261, symmetric pad 6/7
  for (int o = tid; o < 261; o += 128) {
    float lo = 0.f, hi = 0.f;
#pragma unroll
    for (int j = 0; j < 8; ++j) {
      int q = 2 * o + j - 6;
      const int qi = (q < 0) ? (-1 - q) : ((q >= 515) ? (1029 - q) : q);
      const float v = A1[qi];
      lo += c_dec_lo[7 - j] * v;
      hi += c_dec_hi[7 - j] * v;
    }
    A2[o] = lo;
    D2[o] = hi;
  }
  __syncthreads();

  // IDWT level 2 (nc=261 -> 516): la2 = idwt(A2, 0), sa2 = idwt(0, D2)
  for (int m = tid; m < 516; m += 128) {
    float aL = 0.f, aS = 0.f;
#pragma unroll
    for (int j = 0; j < 8; ++j) {
      const int p = m + j - 1;
      if (!(p & 1)) {
        const int t = p >> 1;
        if (t >= 0 && t <= 260) {
          aL += c_dec_lo[j] * A2[t];
          aS += c_dec_hi[j] * D2[t];
        }
      }
    }
    la2[m] = aL;
    sa2[m] = aS;
  }
  __syncthreads();

  // IDWT level 1 (nc=515 -> 1024): la = idwt(la2[:515], 0), sa = idwt(sa2[:515], D1)
  float* Sr = S + ((size_t)b * Cn + c) * Nn;
  float* Lr = L + ((size_t)b * Cn + c) * Nn;
  for (int m = tid; m < Nn; m += 128) {
    float aL = 0.f, aS = 0.f;
#pragma unroll
    for (int j = 0; j < 8; ++j) {
      const int p = m + j - 1;
      if (!(p & 1)) {
        const int t = p >> 1;
        if (t >= 0 && t < 515) {
          aL += c_dec_lo[j] * la2[t];
          aS += c_dec_lo[j] * sa2[t] + c_dec_hi[j] * D1[t];
        }
      }
    }
    Sr[m] = aS;
    Lr[m] = aL;
  }
}

// --- K5: out = G(512x1024) @ [S;L]_b + bias_comb + x  (WMMA GEMM, TDM) ------
__global__ __launch_bounds__(256) void k_gemm_final(
    const float* __restrict__ S, const float* __restrict__ L,
    const float* __restrict__ G, const float* __restrict__ biasc,
    const float* __restrict__ X, float* __restrict__ Out) {
  __shared__ __align__(16) float Alds[128 * LDA];
  __shared__ __align__(16) float Blds[16 * LDB];
  const int tid = threadIdx.x;
  const int lane = tid & 31, wave = tid >> 5;
  const int wave_m = wave & 3, wave_n = wave >> 2;
  const int n0 = blockIdx.x * 128;
  const int m0 = blockIdx.y * 128;
  const int b  = blockIdx.z;
  v8f acc[2][4] = {};
  for (int k0 = 0; k0 < 2 * Cn; k0 += 16) {
    // 16-row B tile lies entirely in S (k0 < 512) or L (k0 >= 512)
    const float* Bsrc = (k0 < Cn) ? (S + ((size_t)b * Cn + k0) * Nn)
                                  : (L + ((size_t)b * Cn + (k0 - Cn)) * Nn);
    if (MS_TDM) {
      if (tid < 32) {
        tdm_load_2d(lds_addr_of(Alds), G + (size_t)m0 * 1024 + k0,
                    16, 128, 1024, Cn, 1024, 3, 3);
        tdm_load_2d(lds_addr_of(Blds), Bsrc + n0,
                    128, 16, Nn, Cn, Nn, 6, 3);
        tdm_wait();
      }
    } else {
#pragma unroll
      for (int i = 0; i < 2; ++i) {  // A tile from G
        const int idx = tid + i * 256;
        const int row = idx >> 2;
        const int c4  = (idx & 3) << 2;
        async_b128(lds_addr_of(&Alds[row * LDA + c4]),
                   G + (size_t)(m0 + row) * 1024 + k0 + c4);
      }
#pragma unroll
      for (int i = 0; i < 2; ++i) {  // B tile from S/L
        const int idx = tid + i * 256;
        const int row = idx >> 5;
        const int c4  = (idx & 31) << 2;
        async_b128(lds_addr_of(&Blds[row * LDB + c4]),
                   Bsrc + (size_t)row * Nn + n0 + c4);
      }
      async_wait();
    }
    __syncthreads();
    wmma_accum_16(Alds, Blds, lane, wave_m, wave_n, acc);
    __syncthreads();
  }
  const int half = lane >> 4, l16 = lane & 15;
  const float* Xb = X + (size_t)b * Cn * Nn;
  float* Ob = Out + (size_t)b * Cn * Nn;
#pragma unroll
  for (int mi = 0; mi < 2; ++mi)
#pragma unroll
    for (int ni = 0; ni < 4; ++ni)
#pragma unroll
      for (int r = 0; r < 8; ++r) {
        const int m = m0 + wave_m * 32 + mi * 16 + r + half * 8;
        const int n = n0 + wave_n * 64 + ni * 16 + l16;
        Ob[(size_t)m * Nn + n] = acc[mi][ni][r] + biasc[m] + Xb[(size_t)m * Nn + n];
      }
}

// ---------------------------------------------------------------------------
extern "C" void kernel_launch(void* const* d_in, const int* in_sizes, int n_in,
                              void* d_out, int out_size, void* d_ws, size_t ws_size,
                              hipStream_t stream) {
  (void)in_sizes; (void)n_in; (void)out_size; (void)ws_size;
  const float* x      = (const float*)d_in[0];
  const float* w_init = (const float*)d_in[1];
  const float* b_init = (const float*)d_in[2];
  const float* w_pf   = (const float*)d_in[3];
  const float* b_pf   = (const float*)d_in[4];
  const float* w_fus  = (const float*)d_in[5];
  const float* b_fus  = (const float*)d_in[6];
  const float *w_b[4], *b_b[4], *g_b[4], *be_b[4];
  for (int i = 0; i < 4; ++i) {
    w_b[i]  = (const float*)d_in[7 + 4 * i];
    b_b[i]  = (const float*)d_in[8 + 4 * i];
    g_b[i]  = (const float*)d_in[9 + 4 * i];
    be_b[i] = (const float*)d_in[10 + 4 * i];
  }
  float* ws = (float*)d_ws;
  const size_t tsz = (size_t)Bn * Cn * Nn;  // 33.5M floats
  float* H = ws;                 // region0: h (dead after branch conv)
  float* S = ws;                 // region0 reused for shorts
  float* Y = ws + tsz;           // region1: branch conv output
  float* L = ws + 2 * tsz;       // region2: longs
  float* G = ws + 3 * tsz;       // 512*1024 combined tail weights
  float* biasc = G + (size_t)Cn * 1024;
  float* stats = biasc + Cn;     // mean[512], inv_std[512]
  float* out = (float*)d_out;

  k_combine<<<dim3(Cn * 1024 / 256), 256, 0, stream>>>(w_fus, w_pf, b_fus, b_pf, G, biasc);
  k_gemm_init<<<dim3(Nn / 128, Cn / 128, Bn), 256, 0, stream>>>(x, w_init, b_init, H);
  k_branch_conv<<<dim3(Nn / 128, 4, Bn), 256, 0, stream>>>(
      H, w_b[0], w_b[1], w_b[2], w_b[3], b_b[0], b_b[1], b_b[2], b_b[3], Y);
  k_stats<<<dim3(Cn), 256, 0, stream>>>(Y, stats);
  k_bn_wavelet<<<dim3(Cn, Bn), 128, 0, stream>>>(
      Y, stats, g_b[0], g_b[1], g_b[2], g_b[3], be_b[0], be_b[1], be_b[2], be_b[3], S, L);
  k_gemm_final<<<dim3(Nn / 128, Cn / 128, Bn), 256, 0, stream>>>(S, L, G, biasc, x, out);
}